// MyGCNConv_72138270704229
// MI455X (gfx1250) — compile-verified
//
#include <hip/hip_runtime.h>

#define DIM 64

typedef __attribute__((ext_vector_type(2))) float v2f;
typedef __attribute__((ext_vector_type(8))) float v8f;

// ---------------------------------------------------------------------------
// Kernel 1: x' = x @ W.T + b  via V_WMMA_F32_16X16X4_F32 (fp32 path, wave32).
// One wave (32 threads) computes a 16x16 output tile; K=64 swept in steps of 4.
// ---------------------------------------------------------------------------
__global__ __launch_bounds__(32)
void gcn_gemm_wmma_f32(const float* __restrict__ X,
                       const float* __restrict__ W,
                       const float* __restrict__ bias,
                       float* __restrict__ XH, int n) {
    const int m0   = blockIdx.x * 16;          // output row tile
    const int j0   = blockIdx.y * 16;          // output col tile
    const int lane = threadIdx.x;              // 0..31
    const int half = lane >> 4;                // 0 | 1
    const int l15  = lane & 15;

    // clamp row for (theoretical) ragged tail; N=100000 is 16-aligned so no divergence
    int arow = m0 + l15; if (arow >= n) arow = n - 1;

    v8f acc = {};
    #pragma unroll
    for (int k0 = 0; k0 < DIM; k0 += 4) {
        // A fragment (16x4): lanes 0-15 -> K={k0,k0+1}, lanes 16-31 -> K={k0+2,k0+3}
        const float* ap = X + (size_t)arow * DIM + k0 + half * 2;
        v2f a; a.x = ap[0]; a.y = ap[1];
        // B fragment (4x16): B[k][j] = W[j0+j][k0+k] (x@W.T), N = l15 per lane
        const float* bp = W + (size_t)(j0 + l15) * DIM + k0 + half * 2;
        v2f bfrag; bfrag.x = bp[0]; bfrag.y = bp[1];
        // 8 args: (neg_a, A, neg_b, B, c_mod, C, reuse_a, reuse_b)
        acc = __builtin_amdgcn_wmma_f32_16x16x4_f32(
            false, a, false, bfrag, (short)0, acc, false, false);
    }

    const float bj = bias[j0 + l15];
    #pragma unroll
    for (int r = 0; r < 8; ++r) {
        const int m = m0 + r + half * 8;       // C/D layout: VGPR r -> M=r / M=8+r
        if (m < n) XH[(size_t)m * DIM + j0 + l15] = acc[r] + bj;
    }
}

// ---------------------------------------------------------------------------
// Kernel 2: initialize degree accumulators to 1.0 (reference: 1 + segment_sum)
// ---------------------------------------------------------------------------
__global__ void gcn_init_deg(float* __restrict__ degi, float* __restrict__ degj, int n) {
    int i = blockIdx.x * blockDim.x + threadIdx.x;
    if (i < n) { degi[i] = 1.0f; degj[i] = 1.0f; }
}

// ---------------------------------------------------------------------------
// Kernel 3: zero the output buffer (used as the atomic aggregation target)
// ---------------------------------------------------------------------------
__global__ void gcn_zero_out(float* __restrict__ out, size_t n) {
    size_t i = (size_t)blockIdx.x * blockDim.x + threadIdx.x;
    if (i < n) out[i] = 0.0f;
}

// ---------------------------------------------------------------------------
// Kernel 4: degree counts via native f32 atomics (L2-resident, 0.4 MB each)
// ---------------------------------------------------------------------------
__global__ void gcn_degree(const long long* __restrict__ ei,
                           float* __restrict__ degi, float* __restrict__ degj, int E) {
    int e = blockIdx.x * blockDim.x + threadIdx.x;
    if (e < E) {
        int r = (int)ei[e];
        int c = (int)ei[(size_t)E + e];
        __hip_atomic_fetch_add(&degi[r], 1.0f, __ATOMIC_RELAXED, __HIP_MEMORY_SCOPE_AGENT);
        __hip_atomic_fetch_add(&degj[c], 1.0f, __ATOMIC_RELAXED, __HIP_MEMORY_SCOPE_AGENT);
    }
}

// ---------------------------------------------------------------------------
// Kernel 5: deg -> deg^-0.5 in place
// ---------------------------------------------------------------------------
__global__ void gcn_rsqrt(float* __restrict__ degi, float* __restrict__ degj, int n) {
    int i = blockIdx.x * blockDim.x + threadIdx.x;
    if (i < n) { degi[i] = rsqrtf(degi[i]); degj[i] = rsqrtf(degj[i]); }
}

// ---------------------------------------------------------------------------
// Kernel 6: edge aggregation. One wave32 per edge; each lane handles a float2
// (lane*2 .. lane*2+1) of the 64-wide row. Gather x'[col] (256B contiguous),
// stream edge_attr, scatter-add into out[row] with native f32 atomics.
// This is the bandwidth-dominant pass (~650 MB total -> ~30us at 23.3 TB/s).
// ---------------------------------------------------------------------------
__global__ __launch_bounds__(256)
void gcn_edge_agg(const float* __restrict__ XH,
                  const float* __restrict__ EA,
                  const long long* __restrict__ ei,
                  const float* __restrict__ dinvi,
                  const float* __restrict__ dinvj,
                  float* __restrict__ out, int E) {
    const size_t tid = (size_t)blockIdx.x * blockDim.x + threadIdx.x;
    const size_t e   = tid >> 5;                 // edge id
    const int lane   = (int)(tid & 31);
    if (e >= (size_t)E) return;

    const int r = (int)ei[e];
    const int c = (int)ei[(size_t)E + e];
    const float val = dinvi[r] * dinvj[c];       // normalized edge weight

    const int d = lane * 2;
    const float2 xv = *(const float2*)(XH + (size_t)c * DIM + d);
    const float2 av = *(const float2*)(EA + e * DIM + d);

    // stream-prefetch the next edge rows of edge_attr into cache
    __builtin_prefetch(EA + (e + 8) * DIM + d, 0, 1);

    float* dst = out + (size_t)r * DIM + d;
    __hip_atomic_fetch_add(dst + 0, val * (xv.x + av.x),
                           __ATOMIC_RELAXED, __HIP_MEMORY_SCOPE_AGENT);
    __hip_atomic_fetch_add(dst + 1, val * (xv.y + av.y),
                           __ATOMIC_RELAXED, __HIP_MEMORY_SCOPE_AGENT);
}

// ---------------------------------------------------------------------------
// Kernel 7: finalize in place: out = relu(agg) + relu(x' + root) * dinvi*dinvj
// ---------------------------------------------------------------------------
__global__ void gcn_finalize(const float* __restrict__ XH,
                             const float* __restrict__ root,
                             const float* __restrict__ dinvi,
                             const float* __restrict__ dinvj,
                             float* __restrict__ out, int n) {
    size_t i = (size_t)blockIdx.x * blockDim.x + threadIdx.x;
    if (i >= (size_t)n * DIM) return;
    const int node = (int)(i / DIM);
    const int d    = (int)(i % DIM);
    const float s  = dinvi[node] * dinvj[node];
    const float h  = XH[i] + root[d];
    out[i] = fmaxf(out[i], 0.0f) + fmaxf(h, 0.0f) * s;
}

// ---------------------------------------------------------------------------
extern "C" void kernel_launch(void* const* d_in, const int* in_sizes, int n_in,
                              void* d_out, int out_size, void* d_ws, size_t ws_size,
                              hipStream_t stream) {
    const float*     x    = (const float*)d_in[0];
    const long long* ei   = (const long long*)d_in[1];   // int64 edge_index [2,E]
    const float*     ea   = (const float*)d_in[2];
    const float*     W    = (const float*)d_in[3];
    const float*     bias = (const float*)d_in[4];
    const float*     root = (const float*)d_in[5];
    float* out = (float*)d_out;

    const int N = in_sizes[0] / DIM;
    const int E = in_sizes[2] / DIM;

    // workspace: x' [N*64] | dinv_i [N] | dinv_j [N]
    float* xh   = (float*)d_ws;
    float* degi = xh + (size_t)N * DIM;
    float* degj = degi + N;

    const size_t nd = (size_t)N * DIM;

    gcn_init_deg<<<(N + 255) / 256, 256, 0, stream>>>(degi, degj, N);
    gcn_zero_out<<<(unsigned)((nd + 255) / 256), 256, 0, stream>>>(out, nd);

    dim3 ggrid((N + 15) / 16, DIM / 16);
    gcn_gemm_wmma_f32<<<ggrid, 32, 0, stream>>>(x, W, bias, xh, N);

    gcn_degree<<<(E + 255) / 256, 256, 0, stream>>>(ei, degi, degj, E);
    gcn_rsqrt<<<(N + 255) / 256, 256, 0, stream>>>(degi, degj, N);

    const size_t work = (size_t)E * 32;
    gcn_edge_agg<<<(unsigned)((work + 255) / 256), 256, 0, stream>>>(
        xh, ea, ei, degi, degj, out, E);

    gcn_finalize<<<(unsigned)((nd + 255) / 256), 256, 0, stream>>>(
        xh, root, degi, degj, out, N);
}